// STU_43447889166478
// MI455X (gfx1250) — compile-verified
//
#include <hip/hip_runtime.h>

// STU layer for MI455X (gfx1250, wave32, WMMA).
//
// Reference math (FFT conv is exact linear conv since n=4096 >= 2*2048-1):
//   U_plus [b,s,k,:] = sum_{t<=s} phi[t,k]          * x[b,s-t,:]
//   U_minus[b,s,k,:] = sum_{t<=s} phi[t,k]*(-1)^t   * x[b,s-t,:]
//   out[b,s,e] = sum_{k,d} U_plus*M_plus[k,d,e] + U_minus*M_minus[k,d,e]
//
// Implemented as WMMA f32 GEMMs (V_WMMA_F32_16X16X4_F32, full f32 precision):
//   stage 1: U_k = P_k * X   (P_k = lower-triangular Toeplitz of filter w_k,
//            A-fragments synthesized from an LDS copy of w_k)  ~140 GFLOP
//   stage 2: out += U_k * M_k  (plus & minus)                  ~ 70 GFLOP
//
// Tiling: 256-thread workgroups (8 waves), 256x128 WG tiles, 64x64 per wave
// (16 accumulators): per K=4 step a wave issues 16 WMMA against 4 A-frags +
// 4 B-frags (~0.75 mem ops per WMMA). Conv launch = 2048 waves, GEMM = 1024.
//
// Workspace: U_plus / U_minus for one k: 2 * (2*2048*512) f32 = 16 MiB.

typedef float v2f __attribute__((ext_vector_type(2)));
typedef float v8f __attribute__((ext_vector_type(8)));

#define SEQ 2048
#define DIM 512
#define NK  16
#define BSZ 2
#define PAD 64                       // zero pad for causal boundary (diff >= -63)
#define UBUF_ELEMS (BSZ * SEQ * DIM) // one variant's U buffer (8 MiB)

// ---------------------------------------------------------------------------
// Stage 1: causal Toeplitz conv  U[var][b][s][d] = sum_r w[s-r] * x[b][r][d]
// grid: x = d-tiles (4 x 128), y = s-tiles (16 x 128), z = (b,var) in 0..3
// block: 256 threads = 8 waves (4x2 of 64x64 wave tiles over 256x128).
// ---------------------------------------------------------------------------
__global__ __launch_bounds__(256)
void stu_conv_kernel(const float* __restrict__ x,
                     const float* __restrict__ phi,
                     float* __restrict__ Ubuf,
                     int k)
{
    __shared__ float w[PAD + SEQ];

    const int z   = blockIdx.z;
    const int b   = z >> 1;
    const int var = z & 1;            // 0: plus, 1: minus (filter *= (-1)^t)
    const int tid = threadIdx.x;

    for (int t = tid; t < PAD; t += 256) w[t] = 0.0f;
    for (int t = tid; t < SEQ; t += 256) {
        float p = phi[t * NK + k];
        if (var && (t & 1)) p = -p;
        w[PAD + t] = p;
    }
    __syncthreads();

    const int wave = tid >> 5;
    const int lane = tid & 31;
    const int half = lane >> 4;       // 0: lanes 0-15, 1: lanes 16-31
    const int l15  = lane & 15;
    const int s_base = blockIdx.y * 128 + (wave >> 1) * 64;
    const int d_base = blockIdx.x * 128 + (wave & 1) * 64;
    const float* X = x + (size_t)b * SEQ * DIM;

    v8f acc[4][4] = {};

    // Causality: rows [s_base, s_base+64) only need r < s_base+64.
    const int r_end = s_base + 64;
    for (int r = 0; r < r_end; r += 4) {
        // A fragments (16x4 f32 Toeplitz tile): A[m,kk] = w[s - (r+kk)]
        // ISA layout: VGPR0 = K(0|2), VGPR1 = K(1|3) for lane halves.
        v2f afr[4];
        #pragma unroll
        for (int sub = 0; sub < 4; ++sub) {
            int diff = s_base + sub * 16 + l15 - r;
            int base = PAD + diff - (half << 1);   // >= 1 by PAD construction
            afr[sub][0] = w[base];
            afr[sub][1] = w[base - 1];
        }
        // B fragments (4x16 f32): B[kk,n] = X[r+kk, col]
        // layout: VGPR0 = K(0|2), VGPR1 = K(1|3) striped over lanes 0-15 / 16-31.
        v2f bfr[4];
        #pragma unroll
        for (int dsub = 0; dsub < 4; ++dsub) {
            int col  = d_base + dsub * 16 + l15;
            int rowK = r + (half << 1);
            bfr[dsub][0] = X[(size_t)rowK * DIM + col];
            bfr[dsub][1] = X[(size_t)(rowK + 1) * DIM + col];
        }
        #pragma unroll
        for (int sub = 0; sub < 4; ++sub)
            #pragma unroll
            for (int dsub = 0; dsub < 4; ++dsub)
                acc[sub][dsub] = __builtin_amdgcn_wmma_f32_16x16x4_f32(
                    false, afr[sub], false, bfr[dsub],
                    (short)0, acc[sub][dsub], false, false);
    }

    // C/D layout: VGPR j -> row j (lanes 0-15) / row j+8 (lanes 16-31)
    float* U = Ubuf + (size_t)var * UBUF_ELEMS + (size_t)b * SEQ * DIM;
    #pragma unroll
    for (int sub = 0; sub < 4; ++sub)
        #pragma unroll
        for (int dsub = 0; dsub < 4; ++dsub) {
            int col = d_base + dsub * 16 + l15;
            #pragma unroll
            for (int j = 0; j < 8; ++j) {
                int row = s_base + sub * 16 + j + (half << 3);
                U[(size_t)row * DIM + col] = acc[sub][dsub][j];
            }
        }
}

// ---------------------------------------------------------------------------
// Stage 2: out[row,e] (+)= U_plus[row,:]*M_plus[k] + U_minus[row,:]*M_minus[k]
// rows = b*2048+s (4096).  grid: x = e-tiles (4 x 128), y = row-tiles (32 x 128)
// block: 256 threads = 8 waves (4x2 of 64x64 wave tiles over 256x128).
// ---------------------------------------------------------------------------
__global__ __launch_bounds__(256)
void stu_gemm_kernel(const float* __restrict__ Ubuf,
                     const float* __restrict__ Mplus,
                     const float* __restrict__ Mminus,
                     float* __restrict__ out,
                     int k, int accumulate)
{
    const int tid  = threadIdx.x;
    const int wave = tid >> 5;
    const int lane = tid & 31;
    const int half = lane >> 4;
    const int l15  = lane & 15;
    const int row_base = blockIdx.y * 128 + (wave >> 1) * 64;
    const int e_base   = blockIdx.x * 128 + (wave & 1) * 64;

    v8f acc[4][4] = {};
    if (accumulate) {
        #pragma unroll
        for (int sub = 0; sub < 4; ++sub)
            #pragma unroll
            for (int dsub = 0; dsub < 4; ++dsub) {
                int col = e_base + dsub * 16 + l15;
                #pragma unroll
                for (int j = 0; j < 8; ++j) {
                    int row = row_base + sub * 16 + j + (half << 3);
                    acc[sub][dsub][j] = out[(size_t)row * DIM + col];
                }
            }
    }

    for (int v = 0; v < 2; ++v) {
        const float* U = Ubuf + (size_t)v * UBUF_ELEMS;
        const float* M = (v ? Mminus : Mplus) + (size_t)k * DIM * DIM;
        for (int d = 0; d < DIM; d += 4) {
            // A[m,kk] = U[rowA, d+kk]: elements (2*half, 2*half+1) are two
            // contiguous, 8-byte-aligned floats -> single b64 load.
            v2f afr[4];
            #pragma unroll
            for (int sub = 0; sub < 4; ++sub) {
                int rowA = row_base + sub * 16 + l15;
                afr[sub] = *(const v2f*)&U[(size_t)rowA * DIM + d + (half << 1)];
            }
            v2f bfr[4];
            #pragma unroll
            for (int dsub = 0; dsub < 4; ++dsub) {
                int col  = e_base + dsub * 16 + l15;
                int rowK = d + (half << 1);
                bfr[dsub][0] = M[(size_t)rowK * DIM + col];
                bfr[dsub][1] = M[(size_t)(rowK + 1) * DIM + col];
            }
            #pragma unroll
            for (int sub = 0; sub < 4; ++sub)
                #pragma unroll
                for (int dsub = 0; dsub < 4; ++dsub)
                    acc[sub][dsub] = __builtin_amdgcn_wmma_f32_16x16x4_f32(
                        false, afr[sub], false, bfr[dsub],
                        (short)0, acc[sub][dsub], false, false);
        }
    }

    #pragma unroll
    for (int sub = 0; sub < 4; ++sub)
        #pragma unroll
        for (int dsub = 0; dsub < 4; ++dsub) {
            int col = e_base + dsub * 16 + l15;
            #pragma unroll
            for (int j = 0; j < 8; ++j) {
                int row = row_base + sub * 16 + j + (half << 3);
                out[(size_t)row * DIM + col] = acc[sub][dsub][j];
            }
        }
}

extern "C" void kernel_launch(void* const* d_in, const int* in_sizes, int n_in,
                              void* d_out, int out_size, void* d_ws, size_t ws_size,
                              hipStream_t stream) {
    const float* x   = (const float*)d_in[0];  // (2, 2048, 512)
    const float* phi = (const float*)d_in[1];  // (2048, 16)
    const float* Mp  = (const float*)d_in[2];  // (16, 512, 512)
    const float* Mm  = (const float*)d_in[3];  // (16, 512, 512)
    // d_in[4] = n (=4096): implicit in the exact linear-conv formulation.
    float* out = (float*)d_out;
    float* U   = (float*)d_ws;                 // needs 2 * 8 MiB = 16 MiB

    (void)in_sizes; (void)n_in; (void)out_size; (void)ws_size;

    for (int k = 0; k < NK; ++k) {
        stu_conv_kernel<<<dim3(4, 16, 4), 256, 0, stream>>>(x, phi, U, k);
        stu_gemm_kernel<<<dim3(4, 32, 1), 256, 0, stream>>>(U, Mp, Mm, out, k, k > 0);
    }
}